// GCNGraphClassifier_86543591014450
// MI455X (gfx1250) — compile-verified
//
#include <hip/hip_runtime.h>

typedef __attribute__((ext_vector_type(2))) float v2f;
typedef __attribute__((ext_vector_type(8))) float v8f;

#define HIDC 16
#define BLK 256

__device__ __forceinline__ void fadd(float* p, float v) {
    __hip_atomic_fetch_add(p, v, __ATOMIC_RELAXED, __HIP_MEMORY_SCOPE_AGENT);
}

// deg=1.0 (self loop), agg=0, pooled=0, cnt=0  (runs every call: graph replay safe)
__global__ void k_init(float* deg, float* agg, float* pooled, float* cnt,
                       int n_nodes, int n_graphs) {
    int i = blockIdx.x * blockDim.x + threadIdx.x;
    int tot = n_nodes * HIDC;
    if (i < tot) agg[i] = 0.0f;
    if (i < n_nodes) deg[i] = 1.0f;
    if (i < n_graphs * HIDC) pooled[i] = 0.0f;
    if (i < n_graphs) cnt[i] = 0.0f;
}

__global__ void k_deg(const int* __restrict__ dst, float* deg, int n_edges) {
    int e = blockIdx.x * blockDim.x + threadIdx.x;
    if (e < n_edges) fadd(&deg[dst[e]], 1.0f);
}

__global__ void k_dinv(float* deg, int n_nodes) {
    int i = blockIdx.x * blockDim.x + threadIdx.x;
    if (i < n_nodes) {
        float d = deg[i];
        deg[i] = (d > 0.0f) ? rsqrtf(d) : 0.0f;
    }
}

__global__ void k_embed(const int* __restrict__ x, const float* __restrict__ emb,
                        float* __restrict__ h, int n_nodes) {
    int i = blockIdx.x * blockDim.x + threadIdx.x;
    if (i < n_nodes * HIDC) {
        int node = i >> 4, j = i & 15;
        h[i] = emb[x[node] * HIDC + j];
    }
}

// hw[tile 16x16] = h[tile 16x16] @ W[16x16], via 4 chained V_WMMA_F32_16X16X4_F32.
// A 16x4 f32 layout: lanes 0-15 -> {K=kk,kk+1} of row M=lane; lanes 16-31 -> {K=kk+2,kk+3}.
// B 4x16 f32 layout (symmetric): lanes 0-15 -> {K=kk,kk+1} of col N=lane; lanes 16-31 -> {kk+2,kk+3}.
// C/D 16x16 f32: VGPR v, lane L -> M = v + 8*(L>=16), N = L&15.
__global__ void k_gcn_wmma(const float* __restrict__ h, const float* __restrict__ W,
                           float* __restrict__ hw, int ntiles) {
    int wave = (blockIdx.x * blockDim.x + threadIdx.x) >> 5;
    int lane = threadIdx.x & 31;
    if (wave >= ntiles) return;            // wave-uniform: EXEC stays all-1s for WMMA
    int half = lane >> 4;                  // 0 or 1
    int l15  = lane & 15;

    v2f bf[4];
#pragma unroll
    for (int t = 0; t < 4; ++t) {
        int kk = 4 * t + 2 * half;
        bf[t][0] = W[(kk + 0) * HIDC + l15];
        bf[t][1] = W[(kk + 1) * HIDC + l15];
    }

    const float* hrow = h + (size_t)(wave * 16 + l15) * HIDC;
    v8f c = {};
#pragma unroll
    for (int t = 0; t < 4; ++t) {
        int kk = 4 * t + 2 * half;
        v2f a;
        a[0] = hrow[kk + 0];
        a[1] = hrow[kk + 1];
        c = __builtin_amdgcn_wmma_f32_16x16x4_f32(
                false, a, false, bf[t], (short)0, c, false, false);
    }

#pragma unroll
    for (int v = 0; v < 8; ++v) {
        int m = v + 8 * half;
        hw[(size_t)(wave * 16 + m) * HIDC + l15] = c[v];
    }
}

// agg[dst] += norm * hw[src], one thread per (edge or self loop)
__global__ void k_scatter(const int* __restrict__ src, const int* __restrict__ dst,
                          const float* __restrict__ dinv, const float* __restrict__ hw,
                          float* __restrict__ agg, int n_edges, int n_nodes) {
    int e = blockIdx.x * blockDim.x + threadIdx.x;
    int total = n_edges + n_nodes;
    if (e >= total) return;
    int s, d;
    if (e < n_edges) { s = src[e]; d = dst[e]; }
    else             { s = e - n_edges; d = s; }
    float nrm = dinv[s] * dinv[d];
    const float4* hr = (const float4*)(hw + (size_t)s * HIDC);
    float* ar = agg + (size_t)d * HIDC;
#pragma unroll
    for (int q = 0; q < 4; ++q) {
        float4 m = hr[q];
        fadd(ar + 4 * q + 0, m.x * nrm);
        fadd(ar + 4 * q + 1, m.y * nrm);
        fadd(ar + 4 * q + 2, m.z * nrm);
        fadd(ar + 4 * q + 3, m.w * nrm);
    }
}

// h = relu(agg + b); agg := 0 (ready for next conv / next launch)
__global__ void k_relu_bias(float* __restrict__ agg, const float* __restrict__ b,
                            float* __restrict__ h, int n_nodes) {
    int i = blockIdx.x * blockDim.x + threadIdx.x;
    if (i < n_nodes * HIDC) {
        int j = i & 15;
        h[i] = fmaxf(agg[i] + b[j], 0.0f);
        agg[i] = 0.0f;
    }
}

__global__ void k_pool(const float* __restrict__ h, const int* __restrict__ batch,
                       float* __restrict__ pooled, float* __restrict__ cnt, int n_nodes) {
    int i = blockIdx.x * blockDim.x + threadIdx.x;
    if (i < n_nodes * HIDC) {
        int node = i >> 4, j = i & 15;
        int g = batch[node];
        fadd(&pooled[g * HIDC + j], h[i]);
        if (j == 0) fadd(&cnt[g], 1.0f);
    }
}

__global__ void k_classify(const float* __restrict__ pooled, const float* __restrict__ cnt,
                           const float* __restrict__ Wc, const float* __restrict__ bc,
                           float* __restrict__ out, int n_graphs, int labels) {
    int i = blockIdx.x * blockDim.x + threadIdx.x;
    if (i >= n_graphs * labels) return;
    int g = i / labels, c = i % labels;
    float inv = 1.0f / fmaxf(cnt[g], 1.0f);
    float acc = bc[c];
#pragma unroll
    for (int j = 0; j < HIDC; ++j)
        acc += pooled[g * HIDC + j] * inv * Wc[j * labels + c];
    out[i] = acc;
}

extern "C" void kernel_launch(void* const* d_in, const int* in_sizes, int n_in,
                              void* d_out, int out_size, void* d_ws, size_t ws_size,
                              hipStream_t stream) {
    const int*   x     = (const int*)d_in[0];
    const int*   ei    = (const int*)d_in[1];
    const int*   batch = (const int*)d_in[2];
    const float* emb   = (const float*)d_in[3];
    const float* W1    = (const float*)d_in[4];
    const float* b1    = (const float*)d_in[5];
    const float* W2    = (const float*)d_in[6];
    const float* b2    = (const float*)d_in[7];
    const float* Wc    = (const float*)d_in[8];
    const float* bc    = (const float*)d_in[9];
    float* out = (float*)d_out;

    const int n_nodes  = in_sizes[0];          // 100000 (divisible by 16)
    const int n_edges  = in_sizes[1] / 2;      // 3200000
    const int labels   = in_sizes[9];          // 10
    const int n_graphs = out_size / labels;    // 512
    const int* src = ei;
    const int* dst = ei + n_edges;

    float* ws     = (float*)d_ws;
    float* deg    = ws;                               // n_nodes  (becomes dinv)
    float* h      = deg + n_nodes;                    // n_nodes*16
    float* hw     = h  + (size_t)n_nodes * HIDC;      // n_nodes*16
    float* agg    = hw + (size_t)n_nodes * HIDC;      // n_nodes*16
    float* pooled = agg + (size_t)n_nodes * HIDC;     // n_graphs*16
    float* cnt    = pooled + (size_t)n_graphs * HIDC; // n_graphs

    const int nh     = n_nodes * HIDC;
    const int ntiles = n_nodes / 16;
    const int wavesPerBlk = BLK / 32;
    const int etotal = n_edges + n_nodes;

    dim3 b(BLK);
    auto g1 = [](int n) { return dim3((n + BLK - 1) / BLK); };
    dim3 gw((ntiles + wavesPerBlk - 1) / wavesPerBlk);

    k_init<<<g1(nh), b, 0, stream>>>(deg, agg, pooled, cnt, n_nodes, n_graphs);
    k_deg<<<g1(n_edges), b, 0, stream>>>(dst, deg, n_edges);
    k_dinv<<<g1(n_nodes), b, 0, stream>>>(deg, n_nodes);
    k_embed<<<g1(nh), b, 0, stream>>>(x, emb, h, n_nodes);

    // conv 1
    k_gcn_wmma<<<gw, b, 0, stream>>>(h, W1, hw, ntiles);
    k_scatter<<<g1(etotal), b, 0, stream>>>(src, dst, deg, hw, agg, n_edges, n_nodes);
    k_relu_bias<<<g1(nh), b, 0, stream>>>(agg, b1, h, n_nodes);

    // conv 2
    k_gcn_wmma<<<gw, b, 0, stream>>>(h, W2, hw, ntiles);
    k_scatter<<<g1(etotal), b, 0, stream>>>(src, dst, deg, hw, agg, n_edges, n_nodes);
    k_relu_bias<<<g1(nh), b, 0, stream>>>(agg, b2, h, n_nodes);

    // pool + classify
    k_pool<<<g1(nh), b, 0, stream>>>(h, batch, pooled, cnt, n_nodes);
    k_classify<<<g1(n_graphs * labels), b, 0, stream>>>(pooled, cnt, Wc, bc, out,
                                                        n_graphs, labels);
}